// CoatnetSelfAttention_58806692216924
// MI455X (gfx1250) — compile-verified
//
#include <hip/hip_runtime.h>

typedef float v2f __attribute__((ext_vector_type(2)));
typedef float v8f __attribute__((ext_vector_type(8)));

#define LOG2E 1.4426950408889634f

__global__ __launch_bounds__(256) void coatnet_attn_kernel(
    const float* __restrict__ x,
    const float* __restrict__ ln_gamma,
    const float* __restrict__ ln_beta,
    const float* __restrict__ qkv_w,   // 8 x 24 row-major
    const float* __restrict__ rab,     // 225 entries
    float* __restrict__ out,
    int nrows)
{
  // LDS: padded strides chosen to dodge bank conflicts / divergence.
  __shared__ float wlds[8 * 32];        // qkv weights, zero-padded cols 24..31
  __shared__ float gblds[16];           // gamma[0..7], beta[8..15]
  __shared__ float blds[15];            // bias[d] * log2e, d = j-i+7
  __shared__ float normlds[256 * 17];   // LayerNorm output, stride 17 (odd)
  __shared__ float qkvlds[256 * 33];    // qkv rows, stride 33 (odd, holds 32 cols)

  const int t   = threadIdx.x;
  const int row = blockIdx.x * 256 + t;
  const bool inb = row < nrows;

  // ---- stage 0: cooperative uniform loads (branch-light, all 256 threads) ----
  {
    const int wr = t >> 5, wc = t & 31;
    wlds[t] = (wc < 24) ? qkv_w[wr * 24 + wc] : 0.f;
  }
  if (t < 16) gblds[t] = (t < 8) ? ln_gamma[t] : ln_beta[t - 8];
  // rel_indices: idx(i,j) = 112 - 14*(j-i); store by d=j-i+7, pre-scaled by log2(e)
  if (t < 15) blds[t] = rab[210 - 14 * t] * LOG2E;

  // ---- stage 1: load x row, LayerNorm ----
  float xr[8];
  if (inb) {
    const float4 a = ((const float4*)(x + (size_t)row * 8))[0];
    const float4 b = ((const float4*)(x + (size_t)row * 8))[1];
    xr[0] = a.x; xr[1] = a.y; xr[2] = a.z; xr[3] = a.w;
    xr[4] = b.x; xr[5] = b.y; xr[6] = b.z; xr[7] = b.w;
  } else {
#pragma unroll
    for (int i = 0; i < 8; ++i) xr[i] = 0.f;
  }

  __syncthreads();  // gblds/wlds/blds visible

  float mu = 0.f;
#pragma unroll
  for (int i = 0; i < 8; ++i) mu += xr[i];
  mu *= 0.125f;
  float var = 0.f;
#pragma unroll
  for (int i = 0; i < 8; ++i) { float d = xr[i] - mu; var = fmaf(d, d, var); }
  var *= 0.125f;
  const float rs = __builtin_amdgcn_rsqf(var + 1e-3f);
#pragma unroll
  for (int i = 0; i < 8; ++i)
    normlds[t * 17 + i] = fmaf((xr[i] - mu) * rs, gblds[i], gblds[8 + i]);

  __syncthreads();

  // ---- stage 2: QKV via V_WMMA_F32_16X16X4_F32 (branch-free) ----
  // A (16x4 f32): lane l%16 = M row; K = j + 2*(l>=16) + 4*chunk
  // B (4x16 f32): lane l%16 = N col; K = j + 2*(l>=16) + 4*chunk
  // D (16x16):    VGPR g, lanes 0-15 -> M=g, lanes 16-31 -> M=g+8; N = l%16
  const int wave = t >> 5;
  const int lane = t & 31;
  const int lm   = lane & 15;
  const int hi   = (lane >> 4) & 1;

  // B fragments are tile-invariant: load once. wr0/wr1 = weight rows for this lane.
  const float* w0 = &wlds[(0 + 2 * hi) * 32];  // K rows {0,1} (+2 if hi)
  const float* w4 = &wlds[(4 + 2 * hi) * 32];  // K rows {4,5} (+2 if hi)
  v2f b00, b01, b10, b11;                      // b[n][chunk]
  b00.x = w0[lm];          b00.y = w0[32 + lm];
  b01.x = w4[lm];          b01.y = w4[32 + lm];
  b10.x = w0[16 + lm];     b10.y = w0[48 + lm];
  b11.x = w4[16 + lm];     b11.y = w4[48 + lm];

#pragma unroll
  for (int tile = 0; tile < 2; ++tile) {
    const int mbase = wave * 32 + tile * 16;
    const float* arow = &normlds[(mbase + lm) * 17];
    v2f a0, a1;
    a0.x = arow[0 + 2 * hi]; a0.y = arow[1 + 2 * hi];   // K chunk 0 (K=0..3)
    a1.x = arow[4 + 2 * hi]; a1.y = arow[5 + 2 * hi];   // K chunk 1 (K=4..7)

    v8f acc0 = {}, acc1 = {};
    acc0 = __builtin_amdgcn_wmma_f32_16x16x4_f32(
        false, a0, false, b00, (short)0, acc0, false, false);
    acc0 = __builtin_amdgcn_wmma_f32_16x16x4_f32(
        false, a1, false, b01, (short)0, acc0, false, false);
    acc1 = __builtin_amdgcn_wmma_f32_16x16x4_f32(
        false, a0, false, b10, (short)0, acc1, false, false);
    acc1 = __builtin_amdgcn_wmma_f32_16x16x4_f32(
        false, a1, false, b11, (short)0, acc1, false, false);

    // Unconditional D stores; cols 24..31 of tile n=1 land in row padding.
    float* qbase = &qkvlds[(mbase + 8 * hi) * 33 + lm];
#pragma unroll
    for (int g = 0; g < 8; ++g) {
      qbase[g * 33]      = acc0[g];
      qbase[g * 33 + 16] = acc1[g];
    }
  }

  __syncthreads();

  // ---- stage 3: per-row attention (rank-1, block-diagonal -> VALU) ----
  const float* myrow = &qkvlds[t * 33];
  float qs[8], ks[8], vs[8], bl[15], accv[8];
#pragma unroll
  for (int j = 0; j < 8; ++j) {
    qs[j] = myrow[j] * LOG2E;   // fold log2(e) so exp == v_exp_f32
    ks[j] = myrow[8 + j];
    vs[j] = myrow[16 + j];
    accv[j] = 0.f;
  }
#pragma unroll
  for (int d = 0; d < 15; ++d) bl[d] = blds[d];

#pragma unroll
  for (int i = 0; i < 8; ++i) {
    float tt[8];
#pragma unroll
    for (int j = 0; j < 8; ++j) tt[j] = fmaf(qs[i], ks[j], bl[j - i + 7]);
    float m = tt[0];
#pragma unroll
    for (int j = 1; j < 8; ++j) m = fmaxf(m, tt[j]);
    float e[8], s = 0.f;
#pragma unroll
    for (int j = 0; j < 8; ++j) { e[j] = __builtin_amdgcn_exp2f(tt[j] - m); s += e[j]; }
    const float sc = vs[i] * __builtin_amdgcn_rcpf(s);  // fold 1/sum into v_i
#pragma unroll
    for (int j = 0; j < 8; ++j) accv[j] = fmaf(sc, e[j], accv[j]);
  }

  if (inb) {
    float4 o0, o1;
    o0.x = xr[0] + accv[0]; o0.y = xr[1] + accv[1];
    o0.z = xr[2] + accv[2]; o0.w = xr[3] + accv[3];
    o1.x = xr[4] + accv[4]; o1.y = xr[5] + accv[5];
    o1.z = xr[6] + accv[6]; o1.w = xr[7] + accv[7];
    ((float4*)(out + (size_t)row * 8))[0] = o0;
    ((float4*)(out + (size_t)row * 8))[1] = o1;
  }
}

extern "C" void kernel_launch(void* const* d_in, const int* in_sizes, int n_in,
                              void* d_out, int out_size, void* d_ws, size_t ws_size,
                              hipStream_t stream) {
  (void)n_in; (void)out_size; (void)d_ws; (void)ws_size;
  const float* x     = (const float*)d_in[0];
  const float* gamma = (const float*)d_in[1];
  const float* beta  = (const float*)d_in[2];
  const float* qkv_w = (const float*)d_in[3];
  const float* rab   = (const float*)d_in[4];
  float* out = (float*)d_out;

  const int nrows = in_sizes[0] / 8;           // 1,048,576
  const int grid  = (nrows + 255) / 256;       // 4096 blocks of 256 (8 waves)
  coatnet_attn_kernel<<<grid, 256, 0, stream>>>(x, gamma, beta, qkv_w, rab, out, nrows);
}